// DCRKT_18511309046071
// MI455X (gfx1250) — compile-verified
//
#include <hip/hip_runtime.h>
#include <hip/hip_bf16.h>
#include <math.h>

#define B_      128
#define NUM_C   1024
#define NUM_Q   10000
#define DQ      128
#define DG      64
#define TOPK    10
#define EDGE_TH 0.05f
#define DECAY   0.5f
#define LN_EPS  1e-5f

typedef __attribute__((ext_vector_type(2))) float v2f;
typedef __attribute__((ext_vector_type(8))) float v8f;

// ---------------------------------------------------------------------------
// Kernel 1: pred (one block per batch element, 128 threads)
// qst MHA block collapses to LN(qt + Wo(Wv qt + bv) + bo) since seq len == 1.
// ---------------------------------------------------------------------------
__global__ void pred_kernel(const int* __restrict__ q_idx,
                            const float* __restrict__ q_emb,
                            const float* __restrict__ wv, const float* __restrict__ bv,
                            const float* __restrict__ wo, const float* __restrict__ bo,
                            const float* __restrict__ lng, const float* __restrict__ lnb,
                            const float* __restrict__ qpw, const float* __restrict__ qpb,
                            const float* __restrict__ mk,
                            const float* __restrict__ mv,
                            float* __restrict__ pred_out)
{
    __shared__ float qt[DQ], vv[DQ], xx[DQ], qh[DQ], pq[DG], mast[DG];
    __shared__ float sim[NUM_C];
    __shared__ float red[DQ];
    __shared__ int   redi[DQ];
    __shared__ float topv[TOPK], attn[TOPK];
    __shared__ int   topi[TOPK];

    const int b = blockIdx.x;
    const int t = threadIdx.x;   // 0..127

    int q = q_idx[b];
    q = min(max(q, 0), NUM_Q - 1);
    qt[t] = q_emb[(size_t)q * DQ + t];
    __syncthreads();

    // v = Wv qt + bv
    {
        float a = bv[t];
        const float* row = wv + (size_t)t * DQ;
        for (int k = 0; k < DQ; ++k) a += row[k] * qt[k];
        vv[t] = a;
    }
    __syncthreads();
    // x = qt + Wo v + bo
    {
        float a = bo[t];
        const float* row = wo + (size_t)t * DQ;
        for (int k = 0; k < DQ; ++k) a += row[k] * vv[k];
        xx[t] = qt[t] + a;
    }
    __syncthreads();

    // LayerNorm
    red[t] = xx[t]; __syncthreads();
    for (int off = 64; off; off >>= 1) { if (t < off) red[t] += red[t + off]; __syncthreads(); }
    const float mu = red[0] * (1.0f / DQ);
    __syncthreads();
    { float d = xx[t] - mu; red[t] = d * d; }
    __syncthreads();
    for (int off = 64; off; off >>= 1) { if (t < off) red[t] += red[t + off]; __syncthreads(); }
    const float var = red[0] * (1.0f / DQ);
    const float rstd = rsqrtf(var + LN_EPS);
    __syncthreads();
    qh[t] = (xx[t] - mu) * rstd * lng[t] + lnb[t];
    __syncthreads();

    // pq = query_proj(qh), then L2-normalize
    if (t < DG) {
        float a = qpb[t];
        const float* row = qpw + (size_t)t * DQ;
        for (int k = 0; k < DQ; ++k) a += row[k] * qh[k];
        pq[t] = a;
    }
    __syncthreads();
    red[t] = (t < DG) ? pq[t] * pq[t] : 0.0f;
    __syncthreads();
    for (int off = 64; off; off >>= 1) { if (t < off) red[t] += red[t + off]; __syncthreads(); }
    const float pinv = 1.0f / fmaxf(sqrtf(red[0]), 1e-12f);
    __syncthreads();
    if (t < DG) pq[t] *= pinv;
    __syncthreads();

    // cosine sim against normalized memory keys
    for (int c = t; c < NUM_C; c += DQ) {
        const float* kr = mk + (size_t)c * DG;
        float dp = 0.0f, nn = 0.0f;
        for (int k = 0; k < DG; ++k) { float km = kr[k]; dp += pq[k] * km; nn += km * km; }
        sim[c] = dp / fmaxf(sqrtf(nn), 1e-12f);
    }
    __syncthreads();

    // top-10 (lowest index wins ties, matching lax.top_k)
    for (int it = 0; it < TOPK; ++it) {
        float bm = -INFINITY; int bi = 0x7fffffff;
        for (int c = t; c < NUM_C; c += DQ) {
            float v = sim[c];
            if (v > bm) { bm = v; bi = c; }
        }
        red[t] = bm; redi[t] = bi;
        __syncthreads();
        for (int off = 64; off; off >>= 1) {
            if (t < off) {
                if (red[t + off] > red[t] ||
                    (red[t + off] == red[t] && redi[t + off] < redi[t])) {
                    red[t] = red[t + off]; redi[t] = redi[t + off];
                }
            }
            __syncthreads();
        }
        if (t == 0) { topv[it] = red[0]; topi[it] = redi[0]; sim[redi[0]] = -INFINITY; }
        __syncthreads();
    }

    if (t == 0) {
        float mx = topv[0], s = 0.0f;
        for (int i = 0; i < TOPK; ++i) { attn[i] = expf(topv[i] - mx); s += attn[i]; }
        float inv = 1.0f / s;
        for (int i = 0; i < TOPK; ++i) attn[i] *= inv;
    }
    __syncthreads();

    if (t < DG) {
        const float* mvb = mv + (size_t)b * NUM_C * DG;
        float a = 0.0f;
        for (int i = 0; i < TOPK; ++i) a += attn[i] * mvb[(size_t)topi[i] * DG + t];
        mast[t] = a;
    }
    __syncthreads();
    red[t] = (t < DG) ? pq[t] * mast[t] : 0.0f;
    __syncthreads();
    for (int off = 64; off; off >>= 1) { if (t < off) red[t] += red[t + off]; __syncthreads(); }
    if (t == 0) pred_out[b] = 1.0f / (1.0f + expf(-red[0]));
}

// ---------------------------------------------------------------------------
// Kernel 2: per-row inverse norms of memory_value (one wave32 per 64-elem row)
// ---------------------------------------------------------------------------
__global__ void rnorm_kernel(const float* __restrict__ mv, float* __restrict__ rinv)
{
    const int t = threadIdx.x;
    const int wave = t >> 5, lane = t & 31;
    const int row = blockIdx.x * 8 + wave;           // 0 .. B*NUM_C-1
    float2 v = *(const float2*)(mv + (size_t)row * DG + lane * 2);
    float s = v.x * v.x + v.y * v.y;
    for (int off = 16; off; off >>= 1) s += __shfl_xor(s, off, 32);
    if (lane == 0) rinv[row] = 1.0f / fmaxf(sqrtf(s), 1e-6f);
}

// ---------------------------------------------------------------------------
// Kernel 3: adj = relu-threshold(normed @ normed^T) + I, batched, WMMA f32.
// 64x64 output tile per 256-thread block; each wave32 owns two 16x16 subtiles.
// K = 64 -> 16 x v_wmma_f32_16x16x4_f32 per subtile.
// ---------------------------------------------------------------------------
__global__ void adj_kernel(const float* __restrict__ mv,
                           const float* __restrict__ rinv,
                           float* __restrict__ adj_out)
{
    __shared__ float As[64][66];   // stride 66: even (8B-aligned float2) + bank skew
    __shared__ float Bs[64][66];

    const int b  = blockIdx.z;
    const int tM = blockIdx.y;
    const int tN = blockIdx.x;
    const int t  = threadIdx.x;    // 0..255

    const float* mvb = mv   + (size_t)b * NUM_C * DG;
    const float* rb  = rinv + (size_t)b * NUM_C;

    // Stage 64x64 A (rows tM*64..) and B (rows tN*64..) into LDS, pre-scaled.
    #pragma unroll
    for (int it = 0; it < 4; ++it) {
        int flat = it * 1024 + t * 4;          // 0..4095
        int r = flat >> 6, k = flat & 63;
        {
            float4 a4 = *(const float4*)(mvb + (size_t)(tM * 64 + r) * DG + k);
            float  sa = rb[tM * 64 + r];
            As[r][k+0] = a4.x * sa; As[r][k+1] = a4.y * sa;
            As[r][k+2] = a4.z * sa; As[r][k+3] = a4.w * sa;
        }
        {
            float4 b4 = *(const float4*)(mvb + (size_t)(tN * 64 + r) * DG + k);
            float  sb = rb[tN * 64 + r];
            Bs[r][k+0] = b4.x * sb; Bs[r][k+1] = b4.y * sb;
            Bs[r][k+2] = b4.z * sb; Bs[r][k+3] = b4.w * sb;
        }
    }
    __syncthreads();

    const int wave = t >> 5;
    const int lane = t & 31;
    const int l16  = lane & 15;
    const int koff = (lane < 16) ? 0 : 2;      // A/B f32 fragment: K{0,1}|K{2,3}

    #pragma unroll
    for (int sub = 0; sub < 2; ++sub) {
        const int s  = wave + sub * 8;         // 16 subtiles in a 4x4 arrangement
        const int sm = (s >> 2) * 16;
        const int sn = (s & 3)  * 16;

        v8f c = {0.f, 0.f, 0.f, 0.f, 0.f, 0.f, 0.f, 0.f};
        #pragma unroll
        for (int k0 = 0; k0 < DG; k0 += 4) {
            v2f a  = *(const v2f*)&As[sm + l16][k0 + koff];
            v2f bb = *(const v2f*)&Bs[sn + l16][k0 + koff];
            c = __builtin_amdgcn_wmma_f32_16x16x4_f32(
                    false, a, false, bb, (short)0, c, false, false);
        }

        // C/D layout: VGPR r -> row (r + (lane>=16 ? 8:0)), col lane&15
        const int gm0 = tM * 64 + sm + ((lane >= 16) ? 8 : 0);
        const int gn  = tN * 64 + sn + l16;
        const size_t base = (size_t)b << 20;   // b * 1024*1024
        #pragma unroll
        for (int r = 0; r < 8; ++r) {
            const int gm = gm0 + r;
            float val = c[r];
            float o = (val > EDGE_TH ? val : 0.0f) + ((gm == gn) ? 1.0f : 0.0f);
            __builtin_nontemporal_store(o, adj_out + base + (size_t)gm * NUM_C + gn);
        }
    }
}

// ---------------------------------------------------------------------------
// Kernel 4: forget gate + blended update (one wave32 per (b,c) row)
// ---------------------------------------------------------------------------
__global__ void update_kernel(const float* __restrict__ mv,
                              const float* __restrict__ ru,
                              const float* __restrict__ dt,
                              const float* __restrict__ fw,
                              const float* __restrict__ fb,
                              float* __restrict__ out)
{
    const int t = threadIdx.x;
    const int wave = t >> 5, lane = t & 31;
    const int row = blockIdx.x * 8 + wave;     // b*NUM_C + c
    const size_t off = (size_t)row * DG + lane * 2;

    float2 m = *(const float2*)(mv + off);
    float2 w = *(const float2*)(fw + lane * 2);
    float s = m.x * w.x + m.y * w.y;
    for (int o = 16; o; o >>= 1) s += __shfl_xor(s, o, 32);

    float tf = log1pf(dt[row]) * DECAY;
    float z  = s + fw[DG] * tf + fb[0];
    float g  = 1.0f / (1.0f + expf(-z));

    float2 r = *(const float2*)(ru + off);
    float2 o2;
    o2.x = g * m.x + (1.0f - g) * r.x;
    o2.y = g * m.y + (1.0f - g) * r.y;
    __builtin_nontemporal_store(o2.x, out + off);
    __builtin_nontemporal_store(o2.y, out + off + 1);
}

// ---------------------------------------------------------------------------
extern "C" void kernel_launch(void* const* d_in, const int* in_sizes, int n_in,
                              void* d_out, int out_size, void* d_ws, size_t ws_size,
                              hipStream_t stream)
{
    (void)in_sizes; (void)n_in; (void)out_size; (void)ws_size;

    // setup_inputs() flatten order: 7 tensors then params in insertion order.
    const int*   q_idx    = (const int*)  d_in[0];
    const float* delta_t  = (const float*)d_in[4];
    const float* resp_upd = (const float*)d_in[5];
    const float* mem_val  = (const float*)d_in[6];
    const float* q_emb    = (const float*)d_in[7];
    // qst block: 31=wq 32=bq 33=wk 34=bk 35=wv 36=bv 37=wo 38=bo 39=ln_g 40=ln_b
    const float* qst_wv   = (const float*)d_in[35];
    const float* qst_bv   = (const float*)d_in[36];
    const float* qst_wo   = (const float*)d_in[37];
    const float* qst_bo   = (const float*)d_in[38];
    const float* qst_lng  = (const float*)d_in[39];
    const float* qst_lnb  = (const float*)d_in[40];
    const float* qpw      = (const float*)d_in[51];
    const float* qpb      = (const float*)d_in[52];
    const float* fw       = (const float*)d_in[53];
    const float* fb       = (const float*)d_in[54];
    const float* mk       = (const float*)d_in[55];

    float* out  = (float*)d_out;
    float* pred = out;                                        // B
    float* adj  = out + B_;                                   // B*NUM_C*NUM_C
    float* upd  = adj + (size_t)B_ * NUM_C * NUM_C;           // B*NUM_C*DG
    float* rinv = (float*)d_ws;                               // B*NUM_C floats

    pred_kernel<<<B_, DQ, 0, stream>>>(q_idx, q_emb, qst_wv, qst_bv, qst_wo, qst_bo,
                                       qst_lng, qst_lnb, qpw, qpb, mk, mem_val, pred);

    rnorm_kernel<<<(B_ * NUM_C) / 8, 256, 0, stream>>>(mem_val, rinv);

    adj_kernel<<<dim3(NUM_C / 64, NUM_C / 64, B_), 256, 0, stream>>>(mem_val, rinv, adj);

    update_kernel<<<(B_ * NUM_C) / 8, 256, 0, stream>>>(mem_val, resp_upd, delta_t,
                                                        fw, fb, upd);
}